// TableMultConsolidate_56865366999600
// MI455X (gfx1250) — compile-verified
//
#include <hip/hip_runtime.h>
#include <stdint.h>

#define EMBED       128
#define CHUNK_ROWS  32
#define THREADS     256     // 8 waves of 32
#define MAX_BLOCKS  1024

typedef float v2f  __attribute__((ext_vector_type(2)));
typedef float v8f  __attribute__((ext_vector_type(8)));
typedef unsigned int u32x4 __attribute__((ext_vector_type(4)));
typedef int  i32x4 __attribute__((ext_vector_type(4)));
typedef int  i32x8 __attribute__((ext_vector_type(8)));

#if __has_builtin(__builtin_amdgcn_tensor_load_to_lds)
#define HAVE_TDM 1
#else
#define HAVE_TDM 0
#endif

#if __has_builtin(__builtin_amdgcn_s_wait_tensorcnt)
#define WAIT_TCNT(n) __builtin_amdgcn_s_wait_tensorcnt(n)
#else
#define WAIT_TCNT(n) asm volatile("s_wait_tensorcnt %0" :: "i"(n) : "memory")
#endif

#if HAVE_TDM
// Build and issue a TDM descriptor for a 2D tile: CHUNK_ROWS x EMBED f32,
// row stride EMBED, from global `gsrc` into LDS byte offset `lds_byte_off`.
__device__ __forceinline__ void tdm_issue(const float* gsrc, unsigned lds_byte_off) {
  unsigned long long ga = (unsigned long long)(uintptr_t)gsrc;
  u32x4 g0;
  g0.x = 1u;                                        // count=1, user descriptor
  g0.y = lds_byte_off;                              // LDS byte address
  g0.z = (unsigned)(ga & 0xFFFFFFFFull);            // global_addr[31:0]
  g0.w = (unsigned)((ga >> 32) & 0x01FFFFFFull)     // global_addr[56:32]
       | (2u << 30);                                // type=2 ("image")

  i32x8 g1;
  g1[0] = (int)(2u << 16);                          // data_size=2 -> 4 bytes
  g1[1] = (int)((unsigned)(EMBED & 0xFFFF) << 16);  // tensor_dim0[15:0]
  g1[2] = (int)(((unsigned)EMBED >> 16)             // tensor_dim0[31:16]
       | ((unsigned)(CHUNK_ROWS & 0xFFFF) << 16));  // tensor_dim1[15:0]
  g1[3] = (int)(((unsigned)CHUNK_ROWS >> 16)        // tensor_dim1[31:16]
       | ((unsigned)EMBED << 16));                  // tile_dim0
  g1[4] = (int)(CHUNK_ROWS & 0xFFFF);               // tile_dim1 (tile_dim2=0 -> 2D)
  g1[5] = EMBED;                                    // tensor_dim0_stride[31:0]
  g1[6] = 0;                                        // stride hi / dim1_stride lo
  g1[7] = 0;

  i32x4 z4 = {0, 0, 0, 0};
#if defined(__clang_major__) && (__clang_major__ >= 23)
  i32x8 z8 = {0, 0, 0, 0, 0, 0, 0, 0};
  __builtin_amdgcn_tensor_load_to_lds(g0, g1, z4, z4, z8, 0);
#else
  __builtin_amdgcn_tensor_load_to_lds(g0, g1, z4, z4, 0);
#endif
}
#endif  // HAVE_TDM

// One 32-row chunk of the GEMV for this wave's 16-column tile.
// A[m][k] = W[base+r+k][d0+m] (from LDS), B[k][n] = x[base+r+k] (bcast),
// D = A*B + C accumulates 16 partial dot products (replicated across columns).
__device__ __forceinline__ v8f wmma_chunk(const float* __restrict__ wbuf,
                                          const float* __restrict__ xg,
                                          int base, int d0, int m, int halfk,
                                          v8f acc) {
#pragma unroll
  for (int r = 0; r < CHUNK_ROWS; r += 4) {
    v2f a;
    a.x = wbuf[(r + halfk) * EMBED + d0 + m];        // K = halfk
    a.y = wbuf[(r + halfk + 1) * EMBED + d0 + m];    // K = halfk+1
    v2f b = *(const v2f*)(xg + base + r + halfk);    // x broadcast across N
    acc = __builtin_amdgcn_wmma_f32_16x16x4_f32(
        false, a, false, b, (short)0, acc, false, false);
  }
  return acc;
}

__global__ __launch_bounds__(THREADS)
void gemv_wmma_kernel(const float* __restrict__ xg, const float* __restrict__ Wg,
                      float* __restrict__ part, int nChunks) {
  __shared__ alignas(16) float sW[2][CHUNK_ROWS * EMBED];  // 2 x 16 KB double buffer

  const int t     = threadIdx.x;
  const int wave  = t >> 5;
  const int lane  = t & 31;
  const int d0    = wave << 4;          // this wave's 16-column tile of D
  const int m     = lane & 15;          // A-matrix row (M)
  const int halfk = (lane >> 4) << 1;   // K base held by this half-wave: 0 or 2

  v8f acc = {};

  const int c0      = (int)blockIdx.x;
  const int cstride = (int)gridDim.x;

#if HAVE_TDM
  const unsigned lds0 = (unsigned)(uintptr_t)&sW[0][0];
  const unsigned lds1 = (unsigned)(uintptr_t)&sW[1][0];

  if (wave == 0 && c0 < nChunks)
    tdm_issue(Wg + (size_t)c0 * CHUNK_ROWS * EMBED, lds0);

  int parity = 0;
  for (int c = c0; c < nChunks; c += cstride) {
    const int nxt = c + cstride;
    if (wave == 0) {
      if (nxt < nChunks) {
        // Prefetch next chunk into the other buffer, then wait for the
        // *current* chunk only (TDM completes in order -> cnt<=1 means
        // the older load finished).
        tdm_issue(Wg + (size_t)nxt * CHUNK_ROWS * EMBED, parity ? lds0 : lds1);
        WAIT_TCNT(1);
      } else {
        WAIT_TCNT(0);
      }
    }
    __syncthreads();                       // chunk c visible to all waves
    acc = wmma_chunk(&sW[parity][0], xg, c * CHUNK_ROWS, d0, m, halfk, acc);
    __syncthreads();                       // all reads done before buffer reuse
    parity ^= 1;
  }
#else
  // Fallback: cooperative synchronous staging (b128 loads + ds stores).
  for (int c = c0; c < nChunks; c += cstride) {
    __syncthreads();
    const float4* src = (const float4*)(Wg + (size_t)c * CHUNK_ROWS * EMBED);
    float4* dst = (float4*)&sW[0][0];
#pragma unroll 4
    for (int i = t; i < (CHUNK_ROWS * EMBED) / 4; i += THREADS) dst[i] = src[i];
    __syncthreads();
    acc = wmma_chunk(&sW[0][0], xg, c * CHUNK_ROWS, d0, m, halfk, acc);
  }
#endif

  // D layout (16x16 f32 C/D): VGPR r, lanes 0-15 -> M=r, lanes 16-31 -> M=r+8.
  // All columns are identical, so lanes 0-7 and 16-23 each hold 8 rows.
  if (m < 8) {
    const int mm = ((lane >> 4) << 3) | m;
    part[(size_t)blockIdx.x * EMBED + d0 + mm] = acc[m];
  }
}

__global__ void reduce_partials_kernel(const float* __restrict__ part,
                                       float* __restrict__ out, int nb) {
  const int d = threadIdx.x;  // 128 threads
  float s = 0.0f;
  for (int b = 0; b < nb; ++b) s += part[(size_t)b * EMBED + d];
  out[d] = s;
}

extern "C" void kernel_launch(void* const* d_in, const int* in_sizes, int n_in,
                              void* d_out, int out_size, void* d_ws, size_t ws_size,
                              hipStream_t stream) {
  (void)n_in; (void)out_size;
  const float* xg = (const float*)d_in[0];   // x: [N] f32
  const float* Wg = (const float*)d_in[1];   // weight: [N,128] f32 (flattened)
  float* out = (float*)d_out;                // [128] f32

  const int N = in_sizes[0];                 // 2,600,000 (divisible by 32)
  const int nChunks = N / CHUNK_ROWS;

  // Each block needs 128 f32 of workspace for its partial; adapt grid to ws.
  int nb = (int)(ws_size / (EMBED * sizeof(float)));
  if (nb > MAX_BLOCKS) nb = MAX_BLOCKS;
  if (nb > nChunks)    nb = nChunks;
  if (nb < 1)          nb = 1;

  float* part = (nb > 1) ? (float*)d_ws : out;
  gemv_wmma_kernel<<<dim3(nb), dim3(THREADS), 0, stream>>>(xg, Wg, part, nChunks);
  if (nb > 1)
    reduce_partials_kernel<<<dim3(1), dim3(EMBED), 0, stream>>>(
        (const float*)d_ws, out, nb);
}